// HungarianMatcher_55018531061939
// MI455X (gfx1250) — compile-verified
//
#include <hip/hip_runtime.h>
#include <stdint.h>

// Problem constants (from reference): B=8, N=K=100, H*W=65536
#define B_ 8
#define N_ 100
#define K_ 100
#define HW_ 65536

// K-split and LDS staging parameters
#define NCHUNK 8               // P split into 8 chunks of 8192
#define PC (HW_ / NCHUNK)      // 8192 P-elements per chunk
#define PSTEP 64               // P-elements staged in LDS per step
#define NSTEPS (PC / PSTEP)    // 128 steps per chunk
#define LDSW 68                // padded LDS row stride (64 + 4) in floats
#define LDS_TILE (32 * LDSW)   // floats per (matrix, buffer)
#define LDS_TILE_BYTES (LDS_TILE * 4)   // 8704 B
#define NBUF 3                 // triple buffer: 2 async steps in flight

typedef __attribute__((ext_vector_type(2))) float v2f;
typedef __attribute__((ext_vector_type(8))) float v8f;

// ---------------------------------------------------------------------------
// Kernel 1: zero the dot-product accumulator (d_out) and the norm scratch.
// ---------------------------------------------------------------------------
__global__ void matcher_zero(float* __restrict__ out, float* __restrict__ norms) {
    int i = blockIdx.x * blockDim.x + threadIdx.x;
    if (i < B_ * N_ * K_) out[i] = 0.0f;
    if (i < 2048)         norms[i] = 0.0f;   // normP[8*128] ++ normG[8*128]
}

// ---------------------------------------------------------------------------
// Kernel 2: batched GEMM  S[b,n,k] = sum_p mp[b,n,p]*mg[b,k,p]  via
// V_WMMA_F32_16X16X4_F32, staged by GLOBAL_LOAD_ASYNC_TO_LDS_B128 with a
// 2-deep async pipeline over 3 LDS buffers: step s computes while steps
// s+1 and s+2 are in flight (ASYNCcnt-tracked, drained 8 at a time).
// Row norms accumulate from the WMMA fragments as packed FMAs (co-execute).
//
// Grid: b(8) x ntb(4) x ktb(4) x chunk(8) = 1024 blocks of 128 threads
// (4 waves; each wave owns one 16x16 subtile of the block's 32x32 tile).
// ---------------------------------------------------------------------------
__global__ __launch_bounds__(128) void matcher_gemm_wmma(
    const float* __restrict__ mp, const float* __restrict__ mg,
    float* __restrict__ outAcc,
    float* __restrict__ normP, float* __restrict__ normG)
{
    __shared__ float ldsA[NBUF][LDS_TILE];
    __shared__ float ldsB[NBUF][LDS_TILE];

    const int tid  = threadIdx.x;
    const int lane = tid & 31;
    const int wave = tid >> 5;

    int blk = blockIdx.x;
    const int chunk = blk & (NCHUNK - 1); blk >>= 3;
    const int ktb   = blk & 3;            blk >>= 2;
    const int ntb   = blk & 3;            blk >>= 2;
    const int b     = blk;                // 0..7

    const int n0 = ntb * 32;
    const int k0 = ktb * 32;
    const int p0 = chunk * PC;

    // cooperative loader mapping: 128 threads cover 32 rows x 64 floats per tile
    const int lrow = tid >> 4;            // 0..7  (rows lrow + 8*i)
    const int lcol = (tid & 15) << 2;     // 0..60 (float4 column)

    // WMMA fragment mapping (ISA 7.12.2, f32 16x16x4):
    //   A vgpr0/1: lanes 0-15 hold K=0/1, lanes 16-31 hold K=2/3, M = lane&15
    //   B mirrors it because B = mg^T (same row-major addressing per lane)
    const int sub_n = (wave & 1) << 4;
    const int sub_k = (wave >> 1) << 4;
    const int ml    = lane & 15;
    const int hi2   = (lane >> 4) << 1;   // 0 or 2

    const float* __restrict__ pA = mp + (size_t)b * N_ * HW_;
    const float* __restrict__ pB = mg + (size_t)b * K_ * HW_;

    // Per-thread async-copy descriptors. Out-of-range rows are clamped to the
    // last valid row: the duplicated data only feeds output elements / norm
    // slots that are never read (guards below), so no masking is needed.
    uint32_t voffA[4], voffB[4];   // 32-bit global byte offsets (GVS mode)
    uint32_t laddrA[4], laddrB[4]; // LDS byte addresses (buffer 0)
    #pragma unroll
    for (int i = 0; i < 4; ++i) {
        const int row = lrow + (i << 3);
        const int gn = n0 + row;
        const int gk = k0 + row;
        const int gnc = gn < N_ ? gn : (N_ - 1);
        const int gkc = gk < K_ ? gk : (K_ - 1);
        voffA[i] = (uint32_t)(((size_t)gnc * HW_ + p0 + lcol) * 4);
        voffB[i] = (uint32_t)(((size_t)gkc * HW_ + p0 + lcol) * 4);
        // Low 32 bits of a flat LDS address == LDS byte offset (ISA aperture map)
        laddrA[i] = (uint32_t)(uintptr_t)&ldsA[0][row * LDSW + lcol];
        laddrB[i] = (uint32_t)(uintptr_t)&ldsB[0][row * LDSW + lcol];
    }

    // Issue the 8 async B128 copies for step s into LDS buffer s%3.
    auto issueStep = [&](int s) {
        const uint32_t ginc = (uint32_t)(s * PSTEP * 4);
        const uint32_t loff = (uint32_t)((s % NBUF) * LDS_TILE_BYTES);
        #pragma unroll
        for (int i = 0; i < 4; ++i) {
            asm volatile("global_load_async_to_lds_b128 %0, %1, %2"
                         :: "v"(laddrA[i] + loff), "v"(voffA[i] + ginc),
                            "s"(pA)
                         : "memory");
            asm volatile("global_load_async_to_lds_b128 %0, %1, %2"
                         :: "v"(laddrB[i] + loff), "v"(voffB[i] + ginc),
                            "s"(pB)
                         : "memory");
        }
    };

    v8f acc = {};
    v2f nA = {0.0f, 0.0f};     // packed row-norm accumulators (v_pk_fma_f32)
    v2f nB = {0.0f, 0.0f};
    // Each A tile is staged identically for every ktb (B for every ntb);
    // waves 0/1 jointly see all 32 A rows, waves 0/2 all 32 B rows.
    const bool accA = (ktb == 0) && (wave < 2);
    const bool accB = (ntb == 0) && ((wave & 1) == 0);

    issueStep(0);              // prime: two steps in flight
    issueStep(1);

    for (int s = 0; s < NSTEPS; ++s) {
        // Async loads complete in order: counter <= 8 means the oldest batch
        // (step s) has fully landed; the step s+1 batch may still be flying.
        if (s + 1 < NSTEPS)
            asm volatile("s_wait_asynccnt 8" ::: "memory");
        else
            asm volatile("s_wait_asynccnt 0" ::: "memory");
        __syncthreads();       // everyone's step-s copies landed; buf (s+2)%3 drained
        if (s + 2 < NSTEPS) issueStep(s + 2);

        const int bs = s % NBUF;
        const float* aRow = &ldsA[bs][(sub_n + ml) * LDSW + hi2];
        const float* bRow = &ldsB[bs][(sub_k + ml) * LDSW + hi2];
        #pragma unroll
        for (int p = 0; p < PSTEP; p += 4) {
            v2f af = *reinterpret_cast<const v2f*>(aRow + p);
            v2f bf = *reinterpret_cast<const v2f*>(bRow + p);
            nA = af * af + nA;     // one v_pk_fma_f32, dual-issues with WMMA
            nB = bf * bf + nB;
            acc = __builtin_amdgcn_wmma_f32_16x16x4_f32(
                /*neg_a=*/false, af, /*neg_b=*/false, bf,
                /*c_mod=*/(short)0, acc, /*reuse_a=*/false, /*reuse_b=*/false);
        }
    }

    // ---- row norms: lane l and l+16 jointly hold row (sub+ml); one atomic/row
    if (accA) {
        float v = nA.x + nA.y;
        v += __shfl_xor(v, 16, 32);
        if (lane < 16)
            atomicAdd(&normP[b * 128 + n0 + sub_n + ml], v);
    }
    if (accB) {
        float v = nB.x + nB.y;
        v += __shfl_xor(v, 16, 32);
        if (lane < 16)
            atomicAdd(&normG[b * 128 + k0 + sub_k + ml], v);
    }

    // ---- scatter partial dot products (C layout: vgpr j -> M=j / j+8, N=lane&15)
    const int gk = k0 + sub_k + ml;
    #pragma unroll
    for (int j = 0; j < 8; ++j) {
        const int gn = n0 + sub_n + j + ((lane >> 4) << 3);
        if (gn < N_ && gk < K_)
            atomicAdd(&outAcc[((size_t)b * N_ + gn) * K_ + gk], acc[j]);
    }
}

// ---------------------------------------------------------------------------
// Kernel 3: out[b,n,k] = term1(y_p,y_gt) * (2*dot) / (||mp_n||^2 + ||mg_k||^2)
// ---------------------------------------------------------------------------
__global__ void matcher_combine(const float* __restrict__ y_p,
                                const float* __restrict__ y_gt,
                                const float* __restrict__ normP,
                                const float* __restrict__ normG,
                                float* __restrict__ out)
{
    int i = blockIdx.x * blockDim.x + threadIdx.x;
    if (i >= B_ * N_ * K_) return;
    const int b = i / (N_ * K_);
    const int r = i - b * (N_ * K_);
    const int n = r / K_;
    const int k = r - n * K_;

    const float yp = y_p[b * N_ + n];
    const float yg = y_gt[b * K_ + k];
    const float t1 = yp * yg + (1.0f - yp) * (1.0f - yg);

    const float num = 2.0f * out[i];
    const float den = normP[b * 128 + n] + normG[b * 128 + k];
    out[i] = t1 * (num / den);
}

// ---------------------------------------------------------------------------
extern "C" void kernel_launch(void* const* d_in, const int* in_sizes, int n_in,
                              void* d_out, int out_size, void* d_ws, size_t ws_size,
                              hipStream_t stream) {
    const float* y_p  = (const float*)d_in[0];   // [8,100]
    const float* y_gt = (const float*)d_in[1];   // [8,100]
    const float* m_p  = (const float*)d_in[2];   // [8,100,256,256]
    const float* m_gt = (const float*)d_in[3];   // [8,100,256,256]
    float* out = (float*)d_out;                  // [8,100,100]

    float* normP = (float*)d_ws;                 // [8*128]
    float* normG = normP + 1024;                 // [8*128]

    const int nOut = B_ * N_ * K_;
    matcher_zero<<<(nOut + 255) / 256, 256, 0, stream>>>(out, normP);
    // b-major block order -> concurrent blocks share one batch's 52 MB (fits L2)
    matcher_gemm_wmma<<<B_ * 4 * 4 * NCHUNK, 128, 0, stream>>>(
        m_p, m_gt, out, normP, normG);
    matcher_combine<<<(nOut + 255) / 256, 256, 0, stream>>>(
        y_p, y_gt, normP, normG, out);
}